// CenterLoss_34583076668114
// MI455X (gfx1250) — compile-verified
//
#include <hip/hip_runtime.h>
#include <hip/hip_bf16.h>

// ---------------------------------------------------------------------------
// CenterLoss on MI455X (gfx1250, wave32).
//   loss = mean( (x - (0.99*center[l] + 0.01*mean_batch[l]))^2 )
// Memory-bound (~100 MB traffic, ~25 MFLOP). Strategy:
//   - hashed-compact scatter (32K slots) instead of 100 MB per-class array
//   - fused loss pass; per-sample SSE computed as the DIAGONAL of the Gram
//     matrix d*d^T via V_WMMA_F32_16X16X4_F32 (free on the matrix pipe).
// ---------------------------------------------------------------------------

#define BATCH   16384
#define FEAT    256
#define NCLASS  100000
#define HASH_SZ 32768
#define HASH_MASK (HASH_SZ - 1)

// workspace layout (bytes)
static constexpr size_t WS_TABLE  = 0;                 // int[32768]   (init 0xFF)
static constexpr size_t WS_COUNTS = 128u << 10;        // int[32768]
static constexpr size_t WS_SLOT   = 256u << 10;        // int[16384]
static constexpr size_t WS_PART   = 320u << 10;        // float[128]
static constexpr size_t WS_SUMS   = 512u << 10;        // float[32768][256]
static constexpr size_t WS_TOTAL  = WS_SUMS + (size_t)HASH_SZ * FEAT * 4;

typedef __attribute__((ext_vector_type(2))) float v2f;
typedef __attribute__((ext_vector_type(8))) float v8f;

#if defined(__HIP_DEVICE_COMPILE__) && __has_builtin(__builtin_amdgcn_wmma_f32_16x16x4_f32)
#define USE_WMMA 1
#else
#define USE_WMMA 0
#endif

__device__ __forceinline__ float wave_sum32(float v) {
#pragma unroll
  for (int o = 16; o > 0; o >>= 1) v += __shfl_xor(v, o, 32);
  return v;
}

// ---------------------------------------------------------------------------
// Pass 1: one wave per sample. Lane 0: hash-insert label -> slot, count++,
// record slot. All lanes: coalesced scatter-add of the 256-f32 feature row
// into sums[slot] (hardware global_atomic_add_f32).
// ---------------------------------------------------------------------------
__global__ void __launch_bounds__(256) cl_build_sums(
    const float* __restrict__ x, const int* __restrict__ labels,
    int* __restrict__ table, int* __restrict__ counts,
    int* __restrict__ slot_of, float* __restrict__ sums) {
  const int wave = (blockIdx.x * blockDim.x + threadIdx.x) >> 5;
  const int lane = threadIdx.x & 31;
  if (wave >= BATCH) return;

  int slot = 0;
  if (lane == 0) {
    const int label = labels[wave];
    unsigned h = ((unsigned)label * 2654435761u) & HASH_MASK;
    for (;;) {
      int prev = atomicCAS(&table[h], -1, label);
      if (prev == -1 || prev == label) break;
      h = (h + 1) & HASH_MASK;
    }
    slot = (int)h;
    atomicAdd(&counts[slot], 1);
    slot_of[wave] = slot;
  }
  slot = __shfl(slot, 0, 32);

  const float* xr = x + (size_t)wave * FEAT;
  float* sr = sums + (size_t)slot * FEAT;
#pragma unroll
  for (int j = 0; j < FEAT / 32; ++j) {
    const int f = lane + j * 32;
    unsafeAtomicAdd(&sr[f], xr[f]);  // global_atomic_add_f32
  }
}

// ---------------------------------------------------------------------------
// Pass 2: one wave per 16-sample tile. Build difference values d[s][k] lane-
// resident in exactly the A(16x4 f32) striping:
//   lanes 0-15 : sample = lane,    K = k, k+1   (VGPR pair)
//   lanes 16-31: sample = lane-16, K = k+2, k+3
// The 4x16 B layout is the lane-transpose of this, so the SAME registers feed
// A and B: WMMA accumulates the Gram d*d^T over K=256; diag(D) = per-sample
// SSE. EXEC is all-1s throughout (full waves, no divergence).
// ---------------------------------------------------------------------------
__global__ void __launch_bounds__(256) cl_loss(
    const float* __restrict__ x, const int* __restrict__ labels,
    const float* __restrict__ center, const int* __restrict__ counts,
    const int* __restrict__ slot_of, const float* __restrict__ sums,
    float* __restrict__ partials) {
  const int lane = threadIdx.x & 31;
  const int wib  = threadIdx.x >> 5;           // 8 waves / block
  const int tile = blockIdx.x * 8 + wib;       // 16 samples / tile
  const int s    = tile * 16 + (lane & 15);

  const int   label = labels[s];
  const int   slot  = slot_of[s];
  const float coef  = 0.01f / (float)counts[slot];   // count >= 1 (present)

  const float* xr = x      + (size_t)s     * FEAT;
  const float* cr = center + (size_t)label * FEAT;
  const float* sr = sums   + (size_t)slot  * FEAT;
  const int koff = (lane >> 4) << 1;           // 0 for lanes 0-15, 2 for 16-31

#if USE_WMMA
  v8f acc = {};
#pragma unroll 4
  for (int k = 0; k < FEAT; k += 4) {
    const v2f xv = *(const v2f*)(xr + k + koff);
    const v2f cv = *(const v2f*)(cr + k + koff);
    const v2f sv = *(const v2f*)(sr + k + koff);
    v2f d;
    d.x = xv.x - 0.99f * cv.x - coef * sv.x;
    d.y = xv.y - 0.99f * cv.y - coef * sv.y;
    // D += d * d^T   (A and B are the same registers; symmetric Gram)
    acc = __builtin_amdgcn_wmma_f32_16x16x4_f32(
        false, d, false, d, (short)0, acc, false, false);
  }
  // Diagonal extraction. D layout: VGPR r, lanes 0-15 -> (M=r, N=lane);
  // lanes 16-31 -> (M=r+8, N=lane-16). Diag lives in lanes 0-7 and 24-31.
  const int m = (lane < 16) ? lane : (lane - 24);
  float pick = 0.0f;
#pragma unroll
  for (int r = 0; r < 8; ++r)
    if (r == m) pick = acc[r];
#else
  // VALU fallback: each (sample, feature) is covered exactly once per wave.
  float pick = 0.0f;
#pragma unroll 4
  for (int k = 0; k < FEAT; k += 4) {
    const v2f xv = *(const v2f*)(xr + k + koff);
    const v2f cv = *(const v2f*)(cr + k + koff);
    const v2f sv = *(const v2f*)(sr + k + koff);
    const float dx = xv.x - 0.99f * cv.x - coef * sv.x;
    const float dy = xv.y - 0.99f * cv.y - coef * sv.y;
    pick = fmaf(dx, dx, pick);
    pick = fmaf(dy, dy, pick);
  }
#endif

  const float wave_sse = wave_sum32(pick);     // tile SSE (16 samples x 256 f)

  __shared__ float sm[8];
  if (lane == 0) sm[wib] = wave_sse;
  __syncthreads();
  if (threadIdx.x == 0) {
    float t = 0.0f;
#pragma unroll
    for (int i = 0; i < 8; ++i) t += sm[i];
    partials[blockIdx.x] = t;
  }
}

// ---------------------------------------------------------------------------
// Pass 3: fold 128 block partials -> scalar mean.
// ---------------------------------------------------------------------------
__global__ void __launch_bounds__(128) cl_final(
    const float* __restrict__ partials, float* __restrict__ out) {
  float v = partials[threadIdx.x];
  v = wave_sum32(v);
  __shared__ float sm[4];
  if ((threadIdx.x & 31) == 0) sm[threadIdx.x >> 5] = v;
  __syncthreads();
  if (threadIdx.x == 0)
    out[0] = (sm[0] + sm[1] + sm[2] + sm[3]) *
             (1.0f / ((float)BATCH * (float)FEAT));
}

// ---------------------------------------------------------------------------
extern "C" void kernel_launch(void* const* d_in, const int* in_sizes, int n_in,
                              void* d_out, int out_size, void* d_ws, size_t ws_size,
                              hipStream_t stream) {
  (void)in_sizes; (void)n_in; (void)out_size; (void)ws_size;
  const float* x      = (const float*)d_in[0];   // [16384, 256] f32
  const int*   labels = (const int*)  d_in[1];   // [16384] int
  const float* center = (const float*)d_in[2];   // [100000, 256] f32
  float* out = (float*)d_out;

  char* ws = (char*)d_ws;
  int*   table    = (int*)  (ws + WS_TABLE);
  int*   counts   = (int*)  (ws + WS_COUNTS);
  int*   slot_of  = (int*)  (ws + WS_SLOT);
  float* partials = (float*)(ws + WS_PART);
  float* sums     = (float*)(ws + WS_SUMS);

  // init: table = -1, everything else = 0 (graph-capturable memset nodes)
  hipMemsetAsync(table, 0xFF, (size_t)HASH_SZ * sizeof(int), stream);
  hipMemsetAsync(ws + WS_COUNTS, 0, WS_TOTAL - WS_COUNTS, stream);

  cl_build_sums<<<BATCH / 8, 256, 0, stream>>>(x, labels, table, counts,
                                               slot_of, sums);
  cl_loss<<<BATCH / 16 / 8, 256, 0, stream>>>(x, labels, center, counts,
                                              slot_of, sums, partials);
  cl_final<<<1, 128, 0, stream>>>(partials, out);
}